// HashGrid4D_22978075034085
// MI455X (gfx1250) — compile-verified
//
#include <hip/hip_runtime.h>
#include <stdint.h>

// HashGrid4D for gfx1250 (MI455X).
// Gather-bound kernel: tables live in the 192MB L2 (RT loads), outputs stream
// to HBM with NT stores, coordinates staged via CDNA5 async global->LDS DMA.

typedef float v4f __attribute__((ext_vector_type(4)));

#define HG_NLEV 8
#define HG_TRES 25

static __device__ __forceinline__ v4f hg_gather2d(const v4f* __restrict__ base,
                                                  unsigned px, unsigned py,
                                                  float fx, float fy, unsigned mask) {
    const unsigned P1 = 2654435761u;
    unsigned hx0 = px;           // PRIMES[0] == 1
    unsigned hx1 = px + 1u;
    unsigned hy0 = py * P1;
    unsigned hy1 = hy0 + P1;
    unsigned i00 = (hx0 ^ hy0) & mask;
    unsigned i01 = (hx0 ^ hy1) & mask;
    unsigned i10 = (hx1 ^ hy0) & mask;
    unsigned i11 = (hx1 ^ hy1) & mask;
    v4f f00 = base[i00];
    v4f f01 = base[i01];
    v4f f10 = base[i10];
    v4f f11 = base[i11];
    float gx = 1.0f - fx, gy = 1.0f - fy;
    v4f r = f00 * (gx * gy);
    r += f01 * (gx * fy);
    r += f10 * (fx * gy);
    r += f11 * (fx * fy);
    return r;
}

static __device__ __forceinline__ v4f hg_gather3d(const v4f* __restrict__ base,
                                                  unsigned px, unsigned py, unsigned pz,
                                                  float fx, float fy, float fz,
                                                  unsigned mask) {
    const unsigned P1 = 2654435761u, P2 = 805459861u;
    unsigned hx0 = px;           // PRIMES[0] == 1
    unsigned hx1 = px + 1u;
    unsigned hy0 = py * P1;
    unsigned hy1 = hy0 + P1;
    unsigned hz0 = pz * P2;
    unsigned hz1 = hz0 + P2;
    unsigned i000 = (hx0 ^ hy0 ^ hz0) & mask;
    unsigned i001 = (hx0 ^ hy0 ^ hz1) & mask;
    unsigned i010 = (hx0 ^ hy1 ^ hz0) & mask;
    unsigned i011 = (hx0 ^ hy1 ^ hz1) & mask;
    unsigned i100 = (hx1 ^ hy0 ^ hz0) & mask;
    unsigned i101 = (hx1 ^ hy0 ^ hz1) & mask;
    unsigned i110 = (hx1 ^ hy1 ^ hz0) & mask;
    unsigned i111 = (hx1 ^ hy1 ^ hz1) & mask;
    v4f f000 = base[i000];
    v4f f001 = base[i001];
    v4f f010 = base[i010];
    v4f f011 = base[i011];
    v4f f100 = base[i100];
    v4f f101 = base[i101];
    v4f f110 = base[i110];
    v4f f111 = base[i111];
    float gx = 1.0f - fx, gy = 1.0f - fy, gz = 1.0f - fz;
    v4f r = f000 * (gx * gy * gz);
    r += f001 * (gx * gy * fz);
    r += f010 * (gx * fy * gz);
    r += f011 * (gx * fy * fz);
    r += f100 * (fx * gy * gz);
    r += f101 * (fx * gy * fz);
    r += f110 * (fx * fy * gz);
    r += f111 * (fx * fy * fz);
    return r;
}

__global__ __launch_bounds__(256)
void HashGrid4D_gfx1250_kernel(const float* __restrict__ x,
                               const float* __restrict__ t,
                               const float* __restrict__ staticT,
                               const float* __restrict__ xyT,
                               const float* __restrict__ xzT,
                               const float* __restrict__ yzT,
                               float* __restrict__ out, int N) {
    __shared__ __align__(16) float sx[256 * 3];
    const int tid = threadIdx.x;
    const long long blk0 = (long long)blockIdx.x * 256;

    // ---- Stage this block's 3KB of coordinates into LDS via CDNA5 async DMA.
    // 192 lanes x 16B fully-coalesced chunks; tail handled with b32 copies.
    {
        const long long totalBytes = (long long)N * 12;
        const long long byte0 = blk0 * 12;
        long long nb = totalBytes - byte0;
        if (nb > 3072) nb = 3072;
        if (tid < 192 && (long long)tid * 16 < nb) {
            const char* g = (const char*)x + byte0 + (long long)tid * 16;
            unsigned lds = (unsigned)(uintptr_t)(&sx[0]) + (unsigned)tid * 16u;
            if ((long long)tid * 16 + 16 <= nb) {
                asm volatile("global_load_async_to_lds_b128 %0, %1, off"
                             :: "v"(lds), "v"(g) : "memory");
            } else {
                int ndw = (int)((nb - (long long)tid * 16) >> 2);
                for (int j = 0; j < ndw; ++j) {
                    asm volatile("global_load_async_to_lds_b32 %0, %1, off"
                                 :: "v"(lds + 4u * (unsigned)j), "v"(g + 4 * j) : "memory");
                }
            }
        }
        asm volatile("s_wait_asynccnt 0x0" ::: "memory");
        __syncthreads();
    }

    const long long n = blk0 + tid;
    if (n >= N) return;

    const float xv = sx[tid * 3 + 0];
    const float yv = sx[tid * 3 + 1];
    const float zv = sx[tid * 3 + 2];

    // Time interpolation setup (uniform for the whole launch; t is a scalar).
    const float t0 = t[0];
    const float ti = t0 * (float)(HG_TRES - 1);
    const float fi = floorf(ti);
    const float tf = ti - fi;
    const int i1 = __builtin_amdgcn_readfirstlane((int)fi);
    const int i2 = __builtin_amdgcn_readfirstlane((int)ceilf(ti));

    const v4f* __restrict__ sB   = (const v4f*)staticT;
    const v4f* __restrict__ xy1B = (const v4f*)xyT + (size_t)i1 * ((size_t)HG_NLEV << 14);
    const v4f* __restrict__ xy2B = (const v4f*)xyT + (size_t)i2 * ((size_t)HG_NLEV << 14);
    const v4f* __restrict__ xz1B = (const v4f*)xzT + (size_t)i1 * ((size_t)HG_NLEV << 12);
    const v4f* __restrict__ xz2B = (const v4f*)xzT + (size_t)i2 * ((size_t)HG_NLEV << 12);
    const v4f* __restrict__ yz1B = (const v4f*)yzT + (size_t)i1 * ((size_t)HG_NLEV << 12);
    const v4f* __restrict__ yz2B = (const v4f*)yzT + (size_t)i2 * ((size_t)HG_NLEV << 12);

    float* __restrict__ outS = out + (size_t)n * 32;
    float* __restrict__ outD = out + (size_t)N * 32 + (size_t)n * 32;

    // floor(512 * (2^(6/7))^l) for l = 0..7
    const float kRes[HG_NLEV] = {512.f, 927.f, 1680.f, 3043.f,
                                 5512.f, 9986.f, 18089.f, 32768.f};
    const float sf = 1.0f - tf;

#pragma unroll
    for (int l = 0; l < HG_NLEV; ++l) {
        const float res = kRes[l];
        float posx = xv * res, posy = yv * res, posz = zv * res;
        float p0x = floorf(posx), p0y = floorf(posy), p0z = floorf(posz);
        float fx = posx - p0x, fy = posy - p0y, fz = posz - p0z;
        unsigned px = (unsigned)p0x, py = (unsigned)p0y, pz = (unsigned)p0z;

        // ---- Static 3D hash grid: 8 corner gathers, trilinear blend.
        v4f fs = hg_gather3d(sB + ((size_t)l << 19), px, py, pz, fx, fy, fz,
                             (1u << 19) - 1u);
        __builtin_nontemporal_store(fs, (v4f*)(outS + l * 4));

        // ---- Dynamic tri-plane: 3 planes x 2 time slices, bilinear + time lerp.
        const unsigned mXY = (1u << 14) - 1u;
        const unsigned mP  = (1u << 12) - 1u;
        v4f a1 = hg_gather2d(xy1B + ((size_t)l << 14), px, py, fx, fy, mXY);
        v4f a2 = hg_gather2d(xy2B + ((size_t)l << 14), px, py, fx, fy, mXY);
        v4f b1 = hg_gather2d(xz1B + ((size_t)l << 12), px, pz, fx, fz, mP);
        v4f b2 = hg_gather2d(xz2B + ((size_t)l << 12), px, pz, fx, fz, mP);
        v4f c1 = hg_gather2d(yz1B + ((size_t)l << 12), py, pz, fy, fz, mP);
        v4f c2 = hg_gather2d(yz2B + ((size_t)l << 12), py, pz, fy, fz, mP);
        v4f va = a1 * sf + a2 * tf;
        v4f vb = b1 * sf + b2 * tf;
        v4f vc = c1 * sf + c2 * tf;
        __builtin_nontemporal_store(va * vb * vc, (v4f*)(outD + l * 4));
    }
}

extern "C" void kernel_launch(void* const* d_in, const int* in_sizes, int n_in,
                              void* d_out, int out_size, void* d_ws, size_t ws_size,
                              hipStream_t stream) {
    (void)n_in; (void)out_size; (void)d_ws; (void)ws_size;
    const float* x  = (const float*)d_in[0];
    const float* t  = (const float*)d_in[1];
    const float* st = (const float*)d_in[2];
    const float* xy = (const float*)d_in[3];
    const float* xz = (const float*)d_in[4];
    const float* yz = (const float*)d_in[5];
    const int N = in_sizes[0] / 3;
    const int blocks = (N + 255) / 256;
    HashGrid4D_gfx1250_kernel<<<blocks, 256, 0, stream>>>(
        x, t, st, xy, xz, yz, (float*)d_out, N);
}